// LocalAttention_73675868996282
// MI455X (gfx1250) — compile-verified
//
#include <hip/hip_runtime.h>
#include <hip/hip_bf16.h>
#include <math.h>

// Problem constants (match reference)
constexpr int BB  = 4;      // batch
constexpr int SS  = 8192;   // seq len
constexpr int CC  = 256;    // channels
constexpr int NWW = 2048;   // words
constexpr int WSS = 256;    // word size (== PSZ == 256)
constexpr int DD  = 32;     // window radius
constexpr int KK  = 2 * DD + 1;  // 65
constexpr int MM  = BB * NWW;    // 8192 rows in the flattened GEMMs

typedef __attribute__((ext_vector_type(2))) float v2f;
typedef __attribute__((ext_vector_type(8))) float v8f;

// ---------------------------------------------------------------------------
// Phase A: for each row m of c_t [MM, 256]:
//   h = tanh(c_t @ W_p^T); logit = h . V_p; p_t = S * sigmoid(logit)
//   u = c_t @ W_a                       (query vector used for scores)
// One block = 16-row M tile. 8 waves x (2 W_p tiles + 2 W_a tiles) cover
// 16 N-tiles x 2 GEMMs. Each tile: K=256 loop of v_wmma_f32_16x16x4_f32.
// The two GEMMs are separate straight-line loops: control flow is wave-
// uniform, so no exec-mask divergence in the inner loops.
// ---------------------------------------------------------------------------
__global__ __launch_bounds__(256) void la_prep_kernel(
    const float* __restrict__ ct,   // [MM, WSS]
    const float* __restrict__ Wa,   // [WSS, CC]   row-major
    const float* __restrict__ Wp,   // [PSZ, WSS]  row-major (we need W_p^T)
    const float* __restrict__ Vp,   // [PSZ]
    float* __restrict__ u,          // [MM, CC]  workspace
    float* __restrict__ pt)         // [MM]      workspace
{
    __shared__ float s_logit[16];

    const int tid  = threadIdx.x;
    const int lane = tid & 31;
    const int wave = tid >> 5;
    const int m0   = blockIdx.x * 16;

    if (tid < 16) s_logit[tid] = 0.0f;
    __syncthreads();

    const int half = lane >> 4;      // lane group: 0 -> K/M low half, 1 -> high
    const int l16  = lane & 15;
    const float* __restrict__ arow = ct + (m0 + l16) * WSS;

    // ---- GEMM 0: h = c_t @ W_p^T, fold tanh(h).V_p into LDS logit ----
#pragma unroll
    for (int rep = 0; rep < 2; ++rep) {
        const int n = (wave + rep * 8) * 16 + l16;
        const float* __restrict__ brow = Wp + n * WSS;   // column n of W_p^T
        v8f acc = {0.f, 0.f, 0.f, 0.f, 0.f, 0.f, 0.f, 0.f};
        for (int k0 = 0; k0 < WSS; k0 += 4) {
            const int kk = k0 + half * 2;
            // A 16x4 frag: VGPR0 = {K=k0 | K=k0+2}, VGPR1 = {K=k0+1 | K=k0+3}
            const v2f a  = *(const v2f*)(arow + kk);
            // B 4x16 frag: contiguous pair along W_p row -> global_load_b64
            const v2f bf = *(const v2f*)(brow + kk);
            acc = __builtin_amdgcn_wmma_f32_16x16x4_f32(
                false, a, false, bf, (short)0, acc, false, false);
        }
        // C/D layout: acc[i] = D[M = i + 8*half][N = n]
        const float vpn = Vp[n];
#pragma unroll
        for (int i = 0; i < 8; ++i)
            atomicAdd(&s_logit[i + half * 8], tanhf(acc[i]) * vpn);
    }

    // ---- GEMM 1: u = c_t @ W_a ----
#pragma unroll
    for (int rep = 0; rep < 2; ++rep) {
        const int n = (wave + rep * 8) * 16 + l16;
        v8f acc = {0.f, 0.f, 0.f, 0.f, 0.f, 0.f, 0.f, 0.f};
        for (int k0 = 0; k0 < WSS; k0 += 4) {
            const int kk = k0 + half * 2;
            const v2f a = *(const v2f*)(arow + kk);
            // B frag rows kk and kk+1 of row-major W_a (coalesced across lanes)
            v2f bf;
            bf.x = Wa[kk * CC + n];
            bf.y = Wa[(kk + 1) * CC + n];
            acc = __builtin_amdgcn_wmma_f32_16x16x4_f32(
                false, a, false, bf, (short)0, acc, false, false);
        }
#pragma unroll
        for (int i = 0; i < 8; ++i)
            u[(m0 + i + half * 8) * CC + n] = acc[i];
    }

    __syncthreads();
    if (tid < 16) {
        const float lg = s_logit[tid];
        pt[m0 + tid] = (float)SS * (1.0f / (1.0f + expf(-lg)));
    }
}

// ---------------------------------------------------------------------------
// Phase B: one block per window (b, w).
//  scores a_k = q[b, si+k-D, :] . u[b,w,:]   (masked -inf out of range)
//  prob = softmax(a) * exp(-2*((s - p)/D)^2)   (Gaussian AFTER normalization)
//  out[b,w,c] = sum_k prob_k * q[b, s_k, c]
// ---------------------------------------------------------------------------
__global__ __launch_bounds__(256) void la_attn_kernel(
    const float* __restrict__ q,    // [BB, SS, CC]
    const float* __restrict__ u,    // [MM, CC]
    const float* __restrict__ pt,   // [MM]
    float* __restrict__ out)        // [MM, CC]
{
    __shared__ float s_u[CC];
    __shared__ float s_score[KK];
    __shared__ float s_prob[KK];
    __shared__ float s_max, s_isum;

    const int bw   = blockIdx.x;
    const int b    = bw >> 11;              // / NWW (2048)
    const int tid  = threadIdx.x;
    const int lane = tid & 31;
    const int wave = tid >> 5;

    const float p  = pt[bw];
    const int   si = (int)p;                // p > 0: trunc == floor == .long()
    const float* __restrict__ qb = q + (size_t)b * SS * CC;

    s_u[tid] = u[(size_t)bw * CC + tid];
    __syncthreads();

    // ---- scores: wave j handles k = j, j+8, ... ----
    // Each lane owns 8 consecutive channels -> two b128 global + two b128 LDS
    // loads per row; wave still covers the contiguous 1KB row.
    const float4* __restrict__ u4 = (const float4*)(s_u + lane * 8);
    const float4  ua = u4[0];
    const float4  ub = u4[1];
    for (int k = wave; k < KK; k += 8) {
        const int s = si + k - DD;
        float val = -INFINITY;
        if (s >= 0 && s < SS) {
            const float4* __restrict__ row4 =
                (const float4*)(qb + (size_t)s * CC + lane * 8);
            const float4 ra = row4[0];
            const float4 rb = row4[1];
            float partial = ra.x * ua.x;
            partial = fmaf(ra.y, ua.y, partial);
            partial = fmaf(ra.z, ua.z, partial);
            partial = fmaf(ra.w, ua.w, partial);
            partial = fmaf(rb.x, ub.x, partial);
            partial = fmaf(rb.y, ub.y, partial);
            partial = fmaf(rb.z, ub.z, partial);
            partial = fmaf(rb.w, ub.w, partial);
#pragma unroll
            for (int off = 16; off > 0; off >>= 1)
                partial += __shfl_xor(partial, off, 32);
            val = partial;
        }
        if (lane == 0) s_score[k] = val;
    }
    __syncthreads();

    // ---- softmax over 65 entries (serial scans are negligible) ----
    if (tid == 0) {
        float m = -INFINITY;
        for (int k = 0; k < KK; ++k) m = fmaxf(m, s_score[k]);
        s_max = m;
    }
    __syncthreads();
    if (tid < KK) {
        const float sc = s_score[tid];
        s_prob[tid] = (sc == -INFINITY) ? 0.0f : expf(sc - s_max);
    }
    __syncthreads();
    if (tid == 0) {
        float sum = 0.0f;
        for (int k = 0; k < KK; ++k) sum += s_prob[k];
        s_isum = 1.0f / sum;
    }
    __syncthreads();
    if (tid < KK) {
        // Gaussian truncation uses integer s (truncated p) minus float p.
        const float sf = (float)(si + tid - DD);
        const float z  = (sf - p) / (float)DD;
        s_prob[tid] *= s_isum * expf(-2.0f * z * z);
    }
    __syncthreads();

    // ---- output: thread <-> channel, coalesced b32 gather of q rows ----
    float accv = 0.0f;
    for (int k = 0; k < KK; ++k) {
        const int s = si + k - DD;
        if (s >= 0 && s < SS)   // invalid rows have prob == 0 (and qw zeroed)
            accv = fmaf(s_prob[k], qb[(size_t)s * CC + tid], accv);
    }
    out[(size_t)bw * CC + tid] = accv;
}

// ---------------------------------------------------------------------------
// Launch. Workspace: u [MM*CC floats] then p_t [MM floats] = 8,421,376 bytes.
// ---------------------------------------------------------------------------
extern "C" void kernel_launch(void* const* d_in, const int* in_sizes, int n_in,
                              void* d_out, int out_size, void* d_ws, size_t ws_size,
                              hipStream_t stream) {
    const float* q  = (const float*)d_in[0];   // [B, S, C]
    const float* ct = (const float*)d_in[1];   // [B, NW, WS]
    const float* Wa = (const float*)d_in[2];   // [WS, C]
    const float* Wp = (const float*)d_in[3];   // [PSZ, WS]
    const float* Vp = (const float*)d_in[4];   // [1, PSZ]

    float* u  = (float*)d_ws;
    float* pt = (float*)((char*)d_ws + (size_t)MM * CC * sizeof(float));
    float* out = (float*)d_out;

    la_prep_kernel<<<MM / 16, 256, 0, stream>>>(ct, Wa, Wp, Vp, u, pt);
    la_attn_kernel<<<MM, 256, 0, stream>>>(q, u, pt, out);
}